// E2R_21328807592128
// MI455X (gfx1250) — compile-verified
//
#include <hip/hip_runtime.h>

typedef __attribute__((ext_vector_type(2))) float v2f;
typedef __attribute__((ext_vector_type(8))) float v8f;

#define LD 65              // padded LDS row stride (bank-conflict free)
#define MS (64 * LD)       // one 64x64 matrix slot, in floats
#define C_DIM 64
#define T_DIM 4096
#define N_EPOCH 8
#define PATCH 512
#define EPS_COV 1e-5f

// ---- block-cooperative 64x64 GEMM in LDS: D = A * B (D disjoint from A,B) ----
// 8 waves, each wave computes two 16x16 tiles via v_wmma_f32_16x16x4_f32.
__device__ __forceinline__ void mm64(float* __restrict__ D, const float* __restrict__ A,
                                     const float* __restrict__ Bm, int lane, int wave) {
    __syncthreads();
#pragma unroll
    for (int i = 0; i < 2; ++i) {
        const int t  = wave + 8 * i;
        const int tm = (t >> 2) * 16;
        const int tn = (t & 3) * 16;
        const int row = tm + (lane & 15);
        const int col = tn + (lane & 15);
        v8f acc = {0.f, 0.f, 0.f, 0.f, 0.f, 0.f, 0.f, 0.f};
#pragma unroll
        for (int k4 = 0; k4 < 64; k4 += 4) {
            const int ko = k4 + ((lane >> 4) << 1);   // K = k4 + {0,2} per half-wave
            v2f a; a.x = A[row * LD + ko];  a.y = A[row * LD + ko + 1];
            v2f b; b.x = Bm[ko * LD + col]; b.y = Bm[(ko + 1) * LD + col];
            acc = __builtin_amdgcn_wmma_f32_16x16x4_f32(false, a, false, b,
                                                        (short)0, acc, false, false);
        }
#pragma unroll
        for (int v = 0; v < 8; ++v) {
            const int m = tm + v + 8 * (lane >> 4);
            D[m * LD + tn + (lane & 15)] = acc[v];
        }
    }
    __syncthreads();
}

#define FOR_EW for (int i = tid; i < 4096; i += 256)

__device__ __forceinline__ void ew_copy(float* d, const float* s, int tid) {
    FOR_EW { int r = i >> 6, c = i & 63; d[r * LD + c] = s[r * LD + c]; }
    __syncthreads();
}
__device__ __forceinline__ void ew_identity(float* d, float dv, int tid) {
    FOR_EW { int r = i >> 6, c = i & 63; d[r * LD + c] = (r == c) ? dv : 0.f; }
    __syncthreads();
}

// ================= kernel 1: per-(b,e) 64x64 covariance via WMMA =================
__global__ void cov64_kernel(const float* __restrict__ x, float* __restrict__ cov) {
    __shared__ float Xs[64 * LD];
    __shared__ float mp[256];
    __shared__ float mean[64];

    const int tid = threadIdx.x, lane = tid & 31, wave = tid >> 5;
    const int bid = blockIdx.x;
    const int b = bid >> 3, e = bid & 7;
    const float* xb = x + (size_t)b * C_DIM * T_DIM + (size_t)e * PATCH;

    float msum = 0.f;
    v8f acc0 = {0.f,0.f,0.f,0.f,0.f,0.f,0.f,0.f};
    v8f acc1 = {0.f,0.f,0.f,0.f,0.f,0.f,0.f,0.f};

    for (int kc = 0; kc < PATCH; kc += 64) {
        __syncthreads();
        for (int i = tid; i < 4096; i += 256) {
            const int c = i >> 6, j = i & 63;
            Xs[c * LD + j] = xb[(size_t)c * T_DIM + kc + j];
        }
        __syncthreads();
        { // per-channel partial sums straight from LDS (no extra HBM traffic)
            const int c = tid & 63, part = tid >> 6;
#pragma unroll
            for (int j = 0; j < 16; ++j) msum += Xs[c * LD + part * 16 + j];
        }
        // G += X * X^T, tiles (wave) and (wave+8)
#pragma unroll
        for (int i = 0; i < 2; ++i) {
            const int t  = wave + 8 * i;
            const int tm = (t >> 2) * 16, tn = (t & 3) * 16;
            const int row = tm + (lane & 15), coln = tn + (lane & 15);
            v8f acc = i ? acc1 : acc0;
#pragma unroll
            for (int k4 = 0; k4 < 64; k4 += 4) {
                const int ko = k4 + ((lane >> 4) << 1);
                v2f a; a.x = Xs[row * LD + ko];  a.y = Xs[row * LD + ko + 1];
                v2f bf; bf.x = Xs[coln * LD + ko]; bf.y = Xs[coln * LD + ko + 1]; // B = X^T
                acc = __builtin_amdgcn_wmma_f32_16x16x4_f32(false, a, false, bf,
                                                            (short)0, acc, false, false);
            }
            if (i) acc1 = acc; else acc0 = acc;
        }
    }
    __syncthreads();
    mp[tid] = msum;
    __syncthreads();
    if (tid < 64)
        mean[tid] = (mp[tid] + mp[tid + 64] + mp[tid + 128] + mp[tid + 192]) * (1.f / PATCH);
    __syncthreads();

    // cov = (G - P * m m^T) / (P-1)
#pragma unroll
    for (int i = 0; i < 2; ++i) {
        const int t  = wave + 8 * i;
        const int tm = (t >> 2) * 16, tn = (t & 3) * 16;
        v8f acc = i ? acc1 : acc0;
#pragma unroll
        for (int v = 0; v < 8; ++v) {
            const int m = tm + v + 8 * (lane >> 4);
            const int n = tn + (lane & 15);
            const float val = (acc[v] - (float)PATCH * mean[m] * mean[n]) * (1.f / (PATCH - 1));
            cov[(size_t)bid * 4096 + m * 64 + n] = val;
        }
    }
}

// ===== kernel 2: normalize -> matrix log (ISS + Newton-Schulz) -> affine -> expm =====
__global__ void spdlogexp_kernel(const float* __restrict__ cov,
                                 const float* __restrict__ alphap,
                                 const float* __restrict__ betap,
                                 float* __restrict__ out) {
    extern __shared__ float lds[];
    float* A  = lds;
    float* Y  = lds + 1 * MS;
    float* Z  = lds + 2 * MS;
    float* Tm = lds + 3 * MS;
    float* W  = lds + 4 * MS;
    float* red = lds + 5 * MS;

    const int tid = threadIdx.x, lane = tid & 31, wave = tid >> 5;
    const int bid = blockIdx.x;
    const float alpha = alphap[0], beta = betap[0];

    for (int i = tid; i < 4096; i += 256)
        A[(i >> 6) * LD + (i & 63)] = cov[(size_t)bid * 4096 + i];
    __syncthreads();
    if (tid < 64) red[tid] = A[tid * LD + tid];
    __syncthreads();
    if (tid == 0) { float s = 0.f; for (int i = 0; i < 64; ++i) s += red[i]; red[0] = 1.f / s; }
    __syncthreads();
    const float invtr = red[0];
    FOR_EW { int r = i >> 6, c = i & 63;
             float v = A[r * LD + c] * invtr;
             if (r == c) v += EPS_COV;
             A[r * LD + c] = v; }
    __syncthreads();

    // ---- logm: 5 Newton-Schulz square roots, then 12-term Mercator series ----
    constexpr int K_SQRT = 5;
    const int ns_it[K_SQRT] = {18, 12, 9, 7, 6};   // slow modes only on the first root
    for (int kq = 0; kq < K_SQRT; ++kq) {
        ew_copy(Y, A, tid);
        ew_identity(Z, 1.f, tid);
        for (int it = 0; it < ns_it[kq]; ++it) {
            mm64(Tm, Z, Y, lane, wave);                 // T = Z*Y
            FOR_EW { int r = i >> 6, c = i & 63;        // T = 1.5 I - 0.5 T
                     float v = Tm[r * LD + c];
                     Tm[r * LD + c] = ((r == c) ? 1.5f : 0.f) - 0.5f * v; }
            __syncthreads();
            mm64(W, Y, Tm, lane, wave);  ew_copy(Y, W, tid);  // Y = Y*T
            mm64(W, Tm, Z, lane, wave);  ew_copy(Z, W, tid);  // Z = T*Z
        }
        ew_copy(A, Y, tid);                              // A <- sqrt(A)
    }
    FOR_EW { int r = i >> 6, c = i & 63;                 // Y = I - A (spectrum <= 0.30)
             Y[r * LD + c] = ((r == c) ? 1.f : 0.f) - A[r * LD + c]; }
    __syncthreads();
    constexpr int LOG_M = 12;                            // log(A) = -(E + E^2/2 + ...)
    ew_identity(Z, 1.f / (float)LOG_M, tid);
    for (int m = LOG_M - 1; m >= 1; --m) {
        mm64(Tm, Y, Z, lane, wave);
        const float dm = 1.f / (float)m;
        FOR_EW { int r = i >> 6, c = i & 63;
                 Z[r * LD + c] = Tm[r * LD + c] + ((r == c) ? dm : 0.f); }
        __syncthreads();
    }
    mm64(Tm, Y, Z, lane, wave);
    const float lscale = -(float)(1 << K_SQRT);          // log(A0) = 2^k * log(A0^(1/2^k))

    // adapted = alpha * log + beta (elementwise), then symmetrize
    FOR_EW { int r = i >> 6, c = i & 63;
             A[r * LD + c] = alpha * (lscale * Tm[r * LD + c]) + beta; }
    __syncthreads();
    FOR_EW { int r = i >> 6, c = i & 63;
             W[r * LD + c] = 0.5f * (A[r * LD + c] + A[c * LD + r]); }
    __syncthreads();

    // ---- expm: scaling-and-squaring, 10-term Taylor via Horner + 6 squarings ----
    constexpr int S_EXP = 6, EXP_M = 10;
    const float es = 1.f / (float)(1 << S_EXP);
    FOR_EW { int r = i >> 6, c = i & 63; Y[r * LD + c] = W[r * LD + c] * es; }
    __syncthreads();
    ew_identity(Z, 1.f, tid);
    for (int m = EXP_M; m >= 1; --m) {
        mm64(Tm, Y, Z, lane, wave);                      // Z = I + (X/m) * Z
        const float dm = 1.f / (float)m;
        FOR_EW { int r = i >> 6, c = i & 63;
                 Z[r * LD + c] = Tm[r * LD + c] * dm + ((r == c) ? 1.f : 0.f); }
        __syncthreads();
    }
    for (int j = 0; j < S_EXP; ++j) {
        mm64(Tm, Z, Z, lane, wave);
        ew_copy(Z, Tm, tid);
    }

    for (int i = tid; i < 4096; i += 256)
        out[(size_t)bid * 4096 + i] = Z[(i >> 6) * LD + (i & 63)];
}

extern "C" void kernel_launch(void* const* d_in, const int* in_sizes, int n_in,
                              void* d_out, int out_size, void* d_ws, size_t ws_size,
                              hipStream_t stream) {
    const float* x     = (const float*)d_in[0];
    const float* alpha = (const float*)d_in[1];
    const float* beta  = (const float*)d_in[2];
    float* out = (float*)d_out;
    float* cov = (float*)d_ws;                    // 2048 * 4096 floats = 32 MB scratch

    const int nmat = 256 * N_EPOCH;               // 2048 workgroups, one per matrix
    cov64_kernel<<<nmat, 256, 0, stream>>>(x, cov);

    const size_t shbytes = (size_t)(5 * MS + 64) * sizeof(float);  // ~83.5 KB LDS
    spdlogexp_kernel<<<nmat, 256, shbytes, stream>>>(cov, alpha, beta, out);
}